// MQuantileLoss_30837865185711
// MI455X (gfx1250) — compile-verified
//
#include <hip/hip_runtime.h>
#include <stdint.h>

// ---------------------------------------------------------------------------
// MQuantileLoss for MI455X (gfx1250)
//  - async global->LDS b128 staging (ASYNCcnt), padded LDS rows (stride 68 fl)
//  - per-thread fp32 cumsum + percentile index via "count of cdf < p"
//  - cdf written back to LDS, Yb/Ya fetched by index
//  - wave reduction via V_WMMA_F32_16X16X4_F32 (B = ones) + one xor-16 shuffle
// ---------------------------------------------------------------------------

#define NBINS          64
#define ROWS_PER_BLOCK 256
#define THREADS        256
#define ROW_STRIDE_F   68      // 68 floats = 272 B: 16B aligned, bank-conflict-free
#define F4_PER_ROW     16      // 64 floats / 4

typedef __attribute__((ext_vector_type(2))) float v2f;
typedef __attribute__((ext_vector_type(8))) float v8f;

// Match the builtin's parameter type exactly (from the round-1 diagnostic):
//   '__attribute__((__vector_size__(4 * sizeof(int)))) int __device__ *'
typedef int v4i_t __attribute__((vector_size(4 * sizeof(int))));
typedef __attribute__((address_space(1))) v4i_t* glb_v4i_p;   // __device__ (AS1)
typedef __attribute__((address_space(3))) v4i_t* lds_v4i_p;   // __shared__ (AS3)

__device__ __forceinline__ void async_copy_b128(const float* g, float* l) {
#if __has_builtin(__builtin_amdgcn_global_load_async_to_lds_b128)
  __builtin_amdgcn_global_load_async_to_lds_b128(
      (glb_v4i_p)g, (lds_v4i_p)l, 0, 0);
#else
#warning "builtin_amdgcn_global_load_async_to_lds_b128 missing -> inline asm path"
  unsigned loff = (unsigned)(uintptr_t)(lds_v4i_p)l;
  asm volatile("global_load_async_to_lds_b128 %0, %1, off"
               :: "v"(loff), "v"((glb_v4i_p)g) : "memory");
#endif
}

__device__ __forceinline__ void wait_async_zero() {
#if __has_builtin(__builtin_amdgcn_s_wait_asynccnt)
  __builtin_amdgcn_s_wait_asynccnt(0);
#else
#warning "builtin_amdgcn_s_wait_asynccnt missing -> inline asm path"
  asm volatile("s_wait_asynccnt 0" ::: "memory");
#endif
}

__global__ void MQ_zero_kernel(float* out) {
  if (threadIdx.x == 0 && blockIdx.x == 0) out[0] = 0.0f;
}

__global__ __launch_bounds__(THREADS)
void MQuantileLoss_kernel(const float* __restrict__ p_est,
                          const float* __restrict__ p_tgt,
                          float* __restrict__ out,
                          int nrows, float inv_scale) {
  __shared__ float lds[ROWS_PER_BLOCK * ROW_STRIDE_F];   // 69,632 B
  __shared__ float wpart[THREADS / 32];

  const int  tid       = threadIdx.x;
  const long tile_row0 = (long)blockIdx.x * ROWS_PER_BLOCK;
  const long myrow     = tile_row0 + tid;
  const bool valid     = myrow < (long)nrows;
  const long total_f4  = (long)nrows * F4_PER_ROW;

  const float PP0 = 0.25f, PP1 = 0.5f, PP2 = 0.75f;
  float q[2][3];

  for (int src = 0; src < 2; ++src) {
    const float* gp = (src == 0) ? p_tgt : p_est;

    // Make sure every wave is done reading LDS before async writes land.
    __syncthreads();

    // Stage the tile: fully coalesced b128 global reads -> padded LDS rows.
    // Flat local float4 index fl = i*THREADS + tid; row = fl>>4, col4 = fl&15.
#pragma unroll
    for (int i = 0; i < F4_PER_ROW; ++i) {
      const int  fl = i * THREADS + tid;
      long f = tile_row0 * F4_PER_ROW + (long)fl;
      if (f >= total_f4) f = total_f4 - 1;          // tail clamp (rows invalid anyway)
      const int r  = fl >> 4;
      const int c4 = fl & 15;
      async_copy_b128(gp + f * 4, &lds[r * ROW_STRIDE_F + c4 * 4]);
    }
    wait_async_zero();
    __syncthreads();

    // Per-thread scan of one row held in LDS (bank-conflict-free b128 reads).
    float*  rowp = &lds[tid * ROW_STRIDE_F];
    float4* row4 = (float4*)rowp;
    float c = 0.0f;
    int n0 = 0, n1 = 0, n2 = 0;
#pragma unroll
    for (int j = 0; j < F4_PER_ROW; ++j) {
      float4 v = row4[j];
      const float a = c + v.x;
      const float b = a + v.y;
      const float d = b + v.z;
      const float e = d + v.w;
      n0 += (a < PP0) + (b < PP0) + (d < PP0) + (e < PP0);
      n1 += (a < PP1) + (b < PP1) + (d < PP1) + (e < PP1);
      n2 += (a < PP2) + (b < PP2) + (d < PP2) + (e < PP2);
      row4[j] = make_float4(a, b, d, e);            // cdf back in place
      c = e;
    }

    const int   idx[3] = {n0, n1, n2};
    const float PP[3]  = {PP0, PP1, PP2};
#pragma unroll
    for (int p = 0; p < 3; ++p) {
      float s = 0.0f;
      const int i = idx[p];
      if (i < NBINS) {                              // searchsorted "found"
        const float Yb = rowp[i];
        const float Ya = (i > 0) ? rowp[i - 1] : 0.0f;
        // Xb - Xa == 1 always; score = Xb - (Yb - p)/(Yb - Ya)
        s = (float)(i + 1) - (Yb - PP[p]) / (Yb - Ya);
      }
      q[src][p] = s;
    }
  }

  float part = 0.0f;
  if (valid) {
    part = fabsf(q[0][0] - q[1][0]) + fabsf(q[0][1] - q[1][1]) +
           fabsf(q[0][2] - q[1][2]);
  }
  part *= inv_scale;

  // ---- Wave reduction on the matrix pipe: D = A(16x4) * ones(4x16) ----
  // A VGPR0: lanes 0-15 -> (M=lane, K=0), lanes 16-31 -> (M=lane-16, K=2).
  // Row-sum S[m] = part[m] + part[m+16], replicated across N.
  // D VGPR r: lanes 0-15 hold S[r], lanes 16-31 hold S[r+8].
  float t;
#if __has_builtin(__builtin_amdgcn_wmma_f32_16x16x4_f32)
  {
    v2f A;  A.x = part;  A.y = 0.0f;
    v2f Bv; Bv.x = 1.0f; Bv.y = 1.0f;
    v8f Cv = {};
    Cv = __builtin_amdgcn_wmma_f32_16x16x4_f32(
        false, A, false, Bv, (short)0, Cv, false, false);
    t = Cv[0] + Cv[1] + Cv[2] + Cv[3] + Cv[4] + Cv[5] + Cv[6] + Cv[7];
    t += __shfl_xor(t, 16, 32);   // S[0..7]+S[8..15] -> full wave sum
  }
#else
#warning "builtin_amdgcn_wmma_f32_16x16x4_f32 missing -> shuffle reduction path"
  t = part;
  for (int d = 16; d >= 1; d >>= 1) t += __shfl_xor(t, d, 32);
#endif

  const int lane = tid & 31, wid = tid >> 5;
  if (lane == 0) wpart[wid] = t;
  __syncthreads();
  if (tid == 0) {
    float s = 0.0f;
#pragma unroll
    for (int w = 0; w < THREADS / 32; ++w) s += wpart[w];
    atomicAdd(out, s);
  }
}

extern "C" void kernel_launch(void* const* d_in, const int* in_sizes, int n_in,
                              void* d_out, int out_size, void* d_ws, size_t ws_size,
                              hipStream_t stream) {
  (void)n_in; (void)out_size; (void)d_ws; (void)ws_size;
  const float* p_est = (const float*)d_in[0];
  const float* p_tgt = (const float*)d_in[1];
  float* out = (float*)d_out;

  const long total = (long)in_sizes[0];
  const int  nrows = (int)(total / NBINS);
  const float inv_scale = 1.0f / (3.0f * (float)nrows);

  MQ_zero_kernel<<<1, 1, 0, stream>>>(out);

  const int nblocks = (nrows + ROWS_PER_BLOCK - 1) / ROWS_PER_BLOCK;
  MQuantileLoss_kernel<<<nblocks, THREADS, 0, stream>>>(p_est, p_tgt, out,
                                                        nrows, inv_scale);
}